// SplineFunction_88570815578839
// MI455X (gfx1250) — compile-verified
//
#include <hip/hip_runtime.h>
#include <stdint.h>

// Piecewise-linear spline evaluation, MI455X (gfx1250).
// B=256, D=256, S=512, NB=32  ->  65536 rows x 512 samples.
// Memory-bound: ~285 MB total traffic, ~12 us floor at 23.3 TB/s.

#define NB 32
#define NK 33            // NB + 1 knots
#define NSAMP 512
#define THREADS 128
#define PER_THREAD (NSAMP / THREADS)   // 4 samples per thread -> B128 load/store

typedef float __attribute__((ext_vector_type(4))) f32x4;

__global__ __launch_bounds__(THREADS) void spline_pl_kernel(
    const float* __restrict__ value,   // [rows, 512]
    const float* __restrict__ xk,      // [rows, 33]
    const float* __restrict__ yk,      // [rows, 33]
    float* __restrict__ out)           // [rows, 512]
{
    __shared__ float kx[NK];    // LDS words 0..32   -> banks 0..32 (conflict-free)
    __shared__ float ky[NK];    // LDS words 33..65
    __shared__ float ks[NB];    // LDS words 66..97  (per-bin slopes)

    const int row = blockIdx.x;
    const int tid = threadIdx.x;

    const float* gx = xk + (size_t)row * NK;
    const float* gy = yk + (size_t)row * NK;

    // --- Stage knots into LDS via gfx1250 async global->LDS DMA ------------
    // GLOBAL_LOAD_ASYNC_TO_LDS_B32: per active lane,
    //   one dword from global memory to the LDS address held in the vdst VGPR; tracked with ASYNCcnt.
    if (tid < NK) {
        unsigned lds_off = (unsigned)(uintptr_t)&kx[tid];
        const float* gp  = gx + tid;
        asm volatile("global_load_async_to_lds_b32 %0, %1, off"
                     :: "v"(lds_off), "v"(gp) : "memory");
    } else if (tid >= 64 && tid < 64 + NK) {
        unsigned lds_off = (unsigned)(uintptr_t)&ky[tid - 64];
        const float* gp  = gy + (tid - 64);
        asm volatile("global_load_async_to_lds_b32 %0, %1, off"
                     :: "v"(lds_off), "v"(gp) : "memory");
    }
    // Each wave drains its own async queue before the workgroup barrier.
    asm volatile("s_wait_asynccnt 0" ::: "memory");
    __syncthreads();

    // --- Precompute per-bin slopes (32 divides per row, not per sample) ----
    if (tid < NB) {
        ks[tid] = (ky[tid + 1] - ky[tid]) / (kx[tid + 1] - kx[tid]);
    }
    __syncthreads();

    // --- Stream 4 samples per thread: NT load -> 5-step search -> NT store -
    const size_t base = (size_t)row * NSAMP + (size_t)tid * PER_THREAD;
    f32x4 v = __builtin_nontemporal_load((const f32x4*)(value + base));
    f32x4 r;
#pragma unroll
    for (int c = 0; c < PER_THREAD; ++c) {
        const float vv = v[c];
        // largest j in [0,31] with kx[j] <= vv  (== clip(searchsorted_right-1,0,31))
        int j = 0;
        j += (vv >= kx[j + 16]) ? 16 : 0;
        j += (vv >= kx[j +  8]) ?  8 : 0;
        j += (vv >= kx[j +  4]) ?  4 : 0;
        j += (vv >= kx[j +  2]) ?  2 : 0;
        j += (vv >= kx[j +  1]) ?  1 : 0;
        r[c] = fmaf(ks[j], vv - kx[j], ky[j]);
    }
    __builtin_nontemporal_store(r, (f32x4*)(out + base));
}

extern "C" void kernel_launch(void* const* d_in, const int* in_sizes, int n_in,
                              void* d_out, int out_size, void* d_ws, size_t ws_size,
                              hipStream_t stream) {
    const float* value = (const float*)d_in[0];
    const float* xk    = (const float*)d_in[1];
    const float* yk    = (const float*)d_in[2];
    float* out         = (float*)d_out;

    const int rows = in_sizes[1] / NK;   // B*D = 65536
    spline_pl_kernel<<<dim3((unsigned)rows), dim3(THREADS), 0, stream>>>(
        value, xk, yk, out);
}